// JointBilateralFilterWithMembership_23545010716706
// MI455X (gfx1250) — compile-verified
//
#include <hip/hip_runtime.h>
#include <stdint.h>

#define HH   256
#define WW   256
#define HWSZ (HH * WW)
#define RAD  5
#define KS   11
#define TILE 16
#define NTHR (TILE * TILE)
#define HALO (TILE + 2 * RAD)   // 26
#define HEL  (HALO * HALO)      // 676
#define NT16 44                 // 44 tiles, one more than ceil(676/16) = 43
#define HELP (NT16 * 16)        // 704: padded so WMMA tiles never go OOB
#define EPSF 1e-8f
// gamma_w = 2.9110 ; z-window equivalent of (0 <= t^2 < gamma^2):
#define ZLO  0.0527766f         // 0.5 / (1 + gamma^2)

typedef float v2f __attribute__((ext_vector_type(2)));
typedef float v8f __attribute__((ext_vector_type(8)));

__device__ __forceinline__ int refl(int i) {
    i = (i < 0) ? -i : i;
    return (i > (HH - 1)) ? (2 * (HH - 1) - i) : i;
}

// CDNA5 async global->LDS b32 copy (ASYNCcnt). Per-lane global address (VGPR
// pair) + per-lane LDS byte address (VGPR).
__device__ __forceinline__ void async_ld_f32(const float* lds_p, const float* g_p) {
    uint32_t loff = (uint32_t)(uintptr_t)lds_p;
    asm volatile("global_load_async_to_lds_b32 %0, %1, off"
                 :: "v"(loff), "v"(g_p)
                 : "memory");
}

__device__ __forceinline__ void wait_async_all() {
    asm volatile("s_wait_asynccnt 0" ::: "memory");
}

__global__ __launch_bounds__(NTHR)
void jbf_membership_kernel(const float* __restrict__ image,
                           const float* __restrict__ guidance,
                           const float* __restrict__ est,
                           const float* __restrict__ var,
                           const float* __restrict__ sinv,
                           float* __restrict__ out)
{
    // Packed LDS planes (u/iq padded to 44 full WMMA tiles):
    //  uA[i] = g0..g3  -> after WMMA transform u0..u3   (ds_load_b128)
    //  uB[i] = g4..g7  -> after WMMA transform u4..u7   (ds_load_b128)
    //  iq[i] = img0, img1, img2, q                      (ds_load_b128)
    //  ev[c][i] = (est_c, var_c)                        (ds_load_b64)
    //  SpL[n][k] = B-tile: (S+S^T)[n][k], rows 8..15 zero-padded
    __shared__ float4 uA[HELP];
    __shared__ float4 uB[HELP];
    __shared__ float4 iq[HELP];
    __shared__ float2 ev[3][HEL];
    __shared__ float  SpL[16 * 8];

    const int tx   = threadIdx.x;
    const int ty   = threadIdx.y;
    const int tid  = ty * TILE + tx;
    const int lane = tid & 31;
    const int x0   = blockIdx.x * TILE;
    const int y0   = blockIdx.y * TILE;
    // wave id, forced into an SGPR so the WMMA tile loop is scalar-controlled
    // (EXEC stays all-ones across the v_wmma ops, as the ISA requires).
    const int wv   = __builtin_amdgcn_readfirstlane(tid >> 5);

    // ---- stage halo via async b32 scatter into packed slots ----
    for (int i = tid; i < HEL; i += NTHR) {
        int ly = i / HALO;
        int lx = i - ly * HALO;
        int gofs = refl(y0 - RAD + ly) * WW + refl(x0 - RAD + lx);
        #pragma unroll
        for (int ch = 0; ch < 4; ++ch)
            async_ld_f32((const float*)&uA[i] + ch, guidance + ch * HWSZ + gofs);
        #pragma unroll
        for (int ch = 4; ch < 8; ++ch)
            async_ld_f32((const float*)&uB[i] + (ch - 4), guidance + ch * HWSZ + gofs);
        #pragma unroll
        for (int c = 0; c < 3; ++c) {
            async_ld_f32((const float*)&ev[c][i] + 0, est + c * HWSZ + gofs);
            async_ld_f32((const float*)&ev[c][i] + 1, var + c * HWSZ + gofs);
            async_ld_f32((const float*)&iq[i] + c, image + c * HWSZ + gofs);
        }
    }

    // Build WMMA B-tile in LDS: SpL[n][k] = sinv[n][k] + sinv[k][n] (n<8), else 0.
    if (tid < 16 * 8) {
        int r = tid >> 3, k = tid & 7;
        SpL[tid] = (r < 8) ? (sinv[r * 8 + k] + sinv[k * 8 + r]) : 0.f;
    }

    wait_async_all();
    __syncthreads();

    // ---- reads of raw data needed before the in-place transform ----
    const int ci = (ty + RAD) * HALO + (tx + RAD);
    const float4 gc0 = uA[ci];
    const float4 gc1 = uB[ci];
    float2 evc[3];
    #pragma unroll
    for (int c = 0; c < 3; ++c) evc[c] = ev[c][ci];

    // Raw g stash for the q pass: fixed 3-slot structure in *named* registers
    // (no dynamic indexing -> no scratch spill). Slots 0/1 always valid.
    const bool own3 = (tid + 2 * NTHR) < HEL;   // tid < 164
    float4 ga0 = uA[tid],         gb0 = uB[tid];
    float4 ga1 = uA[tid + NTHR],  gb1 = uB[tid + NTHR];
    float4 ga2 = {}, gb2 = {};
    if (own3) { ga2 = uA[tid + 2 * NTHR]; gb2 = uB[tid + 2 * NTHR]; }

    // B-matrix registers (4x16 f32, K-pair split by lane>>4, mirroring A layout)
    const int nn = lane & 15;
    const int kb = (lane >> 4) * 2;
    v2f bK0 = *(const v2f*)(SpL + nn * 8 + kb);       // K = kb, kb+1
    v2f bK4 = *(const v2f*)(SpL + nn * 8 + kb + 4);   // K = 4+kb, 4+kb+1

    __syncthreads();   // all raw cross-wave reads done before overwrite

    // ---- in-place GEMM: U(704x8) = G(704x8) * Sp^T via v_wmma_f32_16x16x4_f32.
    // Tiles are wave-private; a wave's A-loads precede its own stores (DS order).
    {
        const int mhalf = (lane >> 4) * 8;
        for (int t = wv; t < NT16; t += 8) {
            int elem = t * 16 + nn;                       // A-row element
            // A 16x4 f32 layout == our float4/element packing: one b64 each.
            v2f a03 = *(const v2f*)((const float*)&uA[0] + elem * 4 + kb);
            v2f a47 = *(const v2f*)((const float*)&uB[0] + elem * 4 + kb);
            v8f acc = {};
            acc = __builtin_amdgcn_wmma_f32_16x16x4_f32(false, a47, false, bK4,
                                                        (short)0, acc, false, false);
            acc = __builtin_amdgcn_wmma_f32_16x16x4_f32(false, a03, false, bK0,
                                                        (short)0, acc, false, false);
            // D: VGPR r holds rows t*16+r (lanes 0-15) / t*16+r+8 (lanes 16-31),
            // column = nn. Scatter columns 0..7 back into the padded uA/uB planes.
            if (nn < 8) {
                float* base = (nn < 4) ? ((float*)&uA[0] + nn)
                                       : ((float*)&uB[0] + (nn - 4));
                int e0 = t * 16 + mhalf;
                #pragma unroll
                for (int r = 0; r < 8; ++r)
                    base[(e0 + r) * 4] = acc[r];
            }
        }
    }
    __syncthreads();

    // ---- q pass: q = 0.5 * dot(g, u) per owned element ----
    {
        float4 u0, u1; float q;
        u0 = uA[tid]; u1 = uB[tid];
        q  = u0.x * ga0.x + u0.y * ga0.y + u0.z * ga0.z + u0.w * ga0.w
           + u1.x * gb0.x + u1.y * gb0.y + u1.z * gb0.z + u1.w * gb0.w;
        ((float*)&iq[tid])[3] = 0.5f * q;

        u0 = uA[tid + NTHR]; u1 = uB[tid + NTHR];
        q  = u0.x * ga1.x + u0.y * ga1.y + u0.z * ga1.z + u0.w * ga1.w
           + u1.x * gb1.x + u1.y * gb1.y + u1.z * gb1.z + u1.w * gb1.w;
        ((float*)&iq[tid + NTHR])[3] = 0.5f * q;

        if (own3) {
            u0 = uA[tid + 2 * NTHR]; u1 = uB[tid + 2 * NTHR];
            q  = u0.x * ga2.x + u0.y * ga2.y + u0.z * ga2.z + u0.w * ga2.w
               + u1.x * gb2.x + u1.y * gb2.y + u1.z * gb2.z + u1.w * gb2.w;
            ((float*)&iq[tid + 2 * NTHR])[3] = 0.5f * q;
        }
    }
    __syncthreads();

    const float qc = iq[ci].w;

    // ---- 11x11 tap loop ----
    float w0 = 0.f, w1 = 0.f, w2 = 0.f, nrm = 0.f;
    for (int dy = 0; dy < KS; ++dy) {
        int rowb = (ty + dy) * HALO + tx;
        #pragma unroll
        for (int dx = 0; dx < KS; ++dx) {
            int li = rowb + dx;
            float4 up0 = uA[li];
            float4 up1 = uB[li];
            float4 iqp = iq[li];

            // mahal = q_p + q_c - dot(g_c, u_p)
            float dot = 0.f;
            dot = fmaf(gc0.x, up0.x, dot);
            dot = fmaf(gc0.y, up0.y, dot);
            dot = fmaf(gc0.z, up0.z, dot);
            dot = fmaf(gc0.w, up0.w, dot);
            dot = fmaf(gc1.x, up1.x, dot);
            dot = fmaf(gc1.y, up1.y, dot);
            dot = fmaf(gc1.z, up1.z, dot);
            dot = fmaf(gc1.w, up1.w, dot);
            float mah = (qc + iqp.w) - dot;
            float bil = __builtin_exp2f(mah * -0.72134752f);   // v_exp_f32

            // membership: t < gamma  <=>  ZLO < z <= 0.5, z = 1 - w + eps
            bool ok = true;
            #pragma unroll
            for (int c = 0; c < 3; ++c) {
                float2 evp = ev[c][li];
                float dd   = evc[c].x - evp.x;
                float d2   = dd * dd;
                float sumv = evc[c].y + evp.y;
                float num  = fmaf(2.f, d2, sumv);
                float den  = fmaxf(num + sumv, EPSF);          // 2*(d2+sumv)
                float w    = num * __builtin_amdgcn_rcpf(den);
                float z    = (1.f + EPSF) - w;
                ok = ok && (z <= 0.5f) && (z > ZLO);
            }
            float wt = ((dy == RAD) && (dx == RAD)) ? 1.f : (ok ? bil : 0.f);

            w0 = fmaf(iqp.x, wt, w0);
            w1 = fmaf(iqp.y, wt, w1);
            w2 = fmaf(iqp.z, wt, w2);
            nrm += wt;
        }
    }

    nrm = fmaxf(nrm, 1e-10f);
    float inv = 1.f / nrm;
    int y = y0 + ty, x = x0 + tx;
    out[(0 * HH + y) * WW + x] = w0 * inv;
    out[(1 * HH + y) * WW + x] = w1 * inv;
    out[(2 * HH + y) * WW + x] = w2 * inv;
}

extern "C" void kernel_launch(void* const* d_in, const int* in_sizes, int n_in,
                              void* d_out, int out_size, void* d_ws, size_t ws_size,
                              hipStream_t stream) {
    (void)in_sizes; (void)n_in; (void)out_size; (void)d_ws; (void)ws_size;
    const float* image    = (const float*)d_in[0];
    const float* guidance = (const float*)d_in[1];
    const float* est      = (const float*)d_in[2];
    const float* var      = (const float*)d_in[3];
    const float* sinv     = (const float*)d_in[4];
    // d_in[5] = spp (unused: gamma_w precomputed for spp=32, as in reference)

    dim3 grid(WW / TILE, HH / TILE);
    dim3 block(TILE, TILE);
    jbf_membership_kernel<<<grid, block, 0, stream>>>(
        image, guidance, est, var, sinv, (float*)d_out);
}